// VDP_MaxPooling_31997506355858
// MI455X (gfx1250) — compile-verified
//
#include <hip/hip_runtime.h>
#include <hip/hip_bf16.h>

// Problem constants (fixed by the reference's setup_inputs)
static constexpr int BB   = 8;    // batch
static constexpr int HH   = 28;   // input H = W
static constexpr int HO   = 14;   // output H = W
static constexpr int CC   = 32;   // channels == wave32 width
static constexpr int NIN  = HH * HH;   // 784
static constexpr int NOUT = HO * HO;   // 196
static constexpr int QG   = 7;         // q values per block (28 * 7 == 196)
static constexpr int MU_N = BB * NOUT * CC;   // 50,176 mu_out elems

// ---------------------------------------------------------------------------
// Phase 1: 2x2 max-pool + argmax index.
// block = (32 c, 14 ox), grid = (14 oy, 8 b)  ->  no integer division at all.
// A wave32 == the 32 channels of one output pixel; all 4 window loads and the
// mu_out / idx stores are fully coalesced.  idx layout: [(b*196+p)*32 + c].
// ---------------------------------------------------------------------------
__global__ void vdp_pool_kernel(const float* __restrict__ mu_in,
                                float* __restrict__ mu_out,
                                int* __restrict__ idx /* may be null */) {
    const int c  = threadIdx.x;      // 0..31
    const int ox = threadIdx.y;      // 0..13
    const int oy = blockIdx.x;       // 0..13
    const int b  = blockIdx.y;       // 0..7
    const int iy = 2 * oy, ix = 2 * ox;
    const int p  = oy * HO + ox;

    const float* base = mu_in + (((size_t)b * HH + iy) * HH + ix) * CC + c;
    const float v00 = base[0];
    const float v01 = base[CC];
    const float v10 = base[HH * CC];
    const float v11 = base[HH * CC + CC];

    // first-match argmax, window order (dy,dx) = (0,0),(0,1),(1,0),(1,1)
    float best = v00; int k = 0;
    if (v01 > best) { best = v01; k = 1; }
    if (v10 > best) { best = v10; k = 2; }
    if (v11 > best) { best = v11; k = 3; }

    const int o = ((b * NOUT) + p) * CC + c;
    mu_out[o] = best;                       // [B,ho,wo,C] flat
    if (idx) {
        idx[o] = (iy + (k >> 1)) * HH + (ix + (k & 1));  // flat input spatial index
    }
}

// ---------------------------------------------------------------------------
// Phase 2: Sigma double-gather.
// block = (32 c, 7 q), grid = (28 q-groups, 196 p, 8 b) -> no division,
// no bounds checks (28*7 == 196 exactly).
// Each Sigma_in cacheline [b,i,j,0..31] (128B) is consumed by exactly one
// wave's single load instruction -> zero cross-wave reuse -> non-temporal
// load/store so the ~630MB stream doesn't evict the hot 200KB index array.
// ---------------------------------------------------------------------------
__global__ void vdp_sigma_gather_kernel(const float* __restrict__ Sigma_in,
                                        const int* __restrict__ idx,
                                        float* __restrict__ Sigma_out) {
    const int c = threadIdx.x;                        // 0..31
    const int q = blockIdx.x * QG + threadIdx.y;      // 0..195
    const int p = blockIdx.y;                         // 0..195
    const int b = blockIdx.z;                         // 0..7

    const int ip = idx[((b * NOUT) + p) * CC + c];    // hot, RT-cached
    const int jq = idx[((b * NOUT) + q) * CC + c];

    const float* src = Sigma_in + (((size_t)b * NIN + ip) * NIN + jq) * CC + c;
    const float v = __builtin_nontemporal_load(src);          // TH=NT: streamed once
    float* dst = Sigma_out + (((size_t)b * NOUT + p) * NOUT + q) * CC + c;
    __builtin_nontemporal_store(v, dst);                      // TH=NT: write-once
}

// ---------------------------------------------------------------------------
// Fallback (only if ws_size is too small): recompute both argmaxes inline.
// mu_in is 0.8MB and stays resident in cache, so the 8 extra loads are cheap.
// ---------------------------------------------------------------------------
__device__ __forceinline__ int vdp_argmax_idx(const float* __restrict__ mu_in,
                                              int b, int p, int c) {
    const int oy = p / HO, ox = p - oy * HO;   // only in fallback path
    const int iy = 2 * oy, ix = 2 * ox;
    const float* base = mu_in + (((size_t)b * HH + iy) * HH + ix) * CC + c;
    float best = base[0]; int k = 0;
    float v;
    v = base[CC];           if (v > best) { best = v; k = 1; }
    v = base[HH * CC];      if (v > best) { best = v; k = 2; }
    v = base[HH * CC + CC]; if (v > best) { best = v; k = 3; }
    return (iy + (k >> 1)) * HH + (ix + (k & 1));
}

__global__ void vdp_sigma_gather_fused_kernel(const float* __restrict__ Sigma_in,
                                              const float* __restrict__ mu_in,
                                              float* __restrict__ Sigma_out) {
    const int c = threadIdx.x;
    const int q = blockIdx.x * QG + threadIdx.y;
    const int p = blockIdx.y;
    const int b = blockIdx.z;

    const int ip = vdp_argmax_idx(mu_in, b, p, c);
    const int jq = vdp_argmax_idx(mu_in, b, q, c);

    const float* src = Sigma_in + (((size_t)b * NIN + ip) * NIN + jq) * CC + c;
    const float v = __builtin_nontemporal_load(src);
    float* dst = Sigma_out + (((size_t)b * NOUT + p) * NOUT + q) * CC + c;
    __builtin_nontemporal_store(v, dst);
}

extern "C" void kernel_launch(void* const* d_in, const int* in_sizes, int n_in,
                              void* d_out, int out_size, void* d_ws, size_t ws_size,
                              hipStream_t stream) {
    (void)in_sizes; (void)n_in; (void)out_size;
    const float* mu_in    = (const float*)d_in[0];
    const float* Sigma_in = (const float*)d_in[1];
    float* mu_out    = (float*)d_out;
    float* Sigma_out = (float*)d_out + MU_N;   // outputs concatenated flat

    const dim3 blk1(CC, HO);            // 32 x 14 = 448 threads, 14 waves
    const dim3 grd1(HO, BB);            // oy, b
    const dim3 blk2(CC, QG);            // 32 x 7 = 224 threads, 7 waves
    const dim3 grd2(NOUT / QG, NOUT, BB);  // 28 q-groups, 196 p, 8 b

    const size_t idx_bytes = (size_t)MU_N * sizeof(int);
    if (ws_size >= idx_bytes) {
        int* idx = (int*)d_ws;
        vdp_pool_kernel<<<grd1, blk1, 0, stream>>>(mu_in, mu_out, idx);
        vdp_sigma_gather_kernel<<<grd2, blk2, 0, stream>>>(Sigma_in, idx, Sigma_out);
    } else {
        vdp_pool_kernel<<<grd1, blk1, 0, stream>>>(mu_in, mu_out, nullptr);
        vdp_sigma_gather_fused_kernel<<<grd2, blk2, 0, stream>>>(Sigma_in, mu_in, Sigma_out);
    }
}